// DeepVCP_35536559407679
// MI455X (gfx1250) — compile-verified
//
#include <hip/hip_runtime.h>

typedef __attribute__((ext_vector_type(2))) float v2f;
typedef __attribute__((ext_vector_type(8))) float v8f;

#define NPTS    2048
#define BATCH   2
#define KKEY    64
#define NSAMP   32
#define NCAND   552
#define KNN     32
#define NQ      (KKEY * NCAND)      /* 35328 = 138 * 256 */
#define QPB     256                 /* queries per block */
#define TCHUNK  64                  /* tgt points per chunk */
#define NCHUNK  (NPTS / TCHUNK)     /* 32 */
#define RADIUS2 1.0f

/* output layout (flat, in tuple order) */
#define OUT_KEYPTS ((size_t)0)
#define OUT_DIST   ((size_t)(BATCH * KKEY * 6))                       /* 768 */
#define OUT_SRCG   (OUT_DIST + (size_t)BATCH * NQ * KNN)              /* 2261760 */
#define OUT_TGTG   (OUT_SRCG + (size_t)BATCH * KKEY * NSAMP * 6)      /* 2286336 */

/* ---------------- kernel 1: per-point MLP score (mean over batch) ------------- */
__global__ void score_kernel(const float* __restrict__ pts,
                             const float* __restrict__ w1, const float* __restrict__ b1,
                             const float* __restrict__ w2, const float* __restrict__ b2,
                             const float* __restrict__ w3, const float* __restrict__ b3,
                             const float* __restrict__ w4, const float* __restrict__ b4,
                             float* __restrict__ scores)
{
    int n = blockIdx.x * blockDim.x + threadIdx.x;
    if (n >= NPTS) return;
    float acc = 0.f;
    for (int b = 0; b < BATCH; ++b) {
        float x[6];
#pragma unroll
        for (int c = 0; c < 6; ++c) x[c] = pts[((size_t)b * 6 + c) * NPTS + n];
        float feat[32];
#pragma unroll
        for (int o2 = 0; o2 < 32; ++o2) feat[o2] = b2[o2];
        for (int o = 0; o < 64; ++o) {
            float h = b1[o];
#pragma unroll
            for (int c = 0; c < 6; ++c) h += w1[o * 6 + c] * x[c];
            h = fmaxf(h, 0.f);
#pragma unroll
            for (int o2 = 0; o2 < 32; ++o2) feat[o2] += w2[o2 * 64 + o] * h;
        }
        float s = b4[0];
        for (int o3 = 0; o3 < 16; ++o3) {
            float g = b3[o3];
#pragma unroll
            for (int o2 = 0; o2 < 32; ++o2) g += w3[o3 * 32 + o2] * feat[o2];
            s += w4[o3] * fmaxf(g, 0.f);
        }
        acc += s;
    }
    scores[n] = acc * (1.0f / BATCH);
}

/* --------- kernel 2: top-64 (desc, tie->low idx) + keypoint gather ------------ */
__global__ void topk_gather_kernel(const float* __restrict__ pts,
                                   const float* __restrict__ scores,
                                   float* __restrict__ keypts,       /* [B][64][6] */
                                   float* __restrict__ dout_keypts)  /* or null */
{
    __shared__ float sc[NPTS];
    __shared__ float rv[256];
    __shared__ int   ri[256];
    __shared__ int   kidx[KKEY];
    int tid = threadIdx.x;
    for (int i = tid; i < NPTS; i += 256) sc[i] = scores[i];
    __syncthreads();
    for (int r = 0; r < KKEY; ++r) {
        float bv = -__builtin_inff(); int bi = NPTS;
        for (int i = tid; i < NPTS; i += 256) {
            float v = sc[i];
            if (v > bv || (v == bv && i < bi)) { bv = v; bi = i; }
        }
        rv[tid] = bv; ri[tid] = bi;
        __syncthreads();
        for (int off = 128; off > 0; off >>= 1) {
            if (tid < off) {
                float ov = rv[tid + off]; int oi = ri[tid + off];
                if (ov > rv[tid] || (ov == rv[tid] && oi < ri[tid])) { rv[tid] = ov; ri[tid] = oi; }
            }
            __syncthreads();
        }
        if (tid == 0) { kidx[r] = ri[0]; sc[ri[0]] = -__builtin_inff(); }
        __syncthreads();
    }
    for (int i = tid; i < BATCH * KKEY * 6; i += 256) {
        int c = i % 6, k = (i / 6) % KKEY, b = i / (6 * KKEY);
        float v = pts[((size_t)b * 6 + c) * NPTS + kidx[k]];
        keypts[i] = v;
        if (dout_keypts) dout_keypts[i] = v;
    }
}

/* ------------- kernel 3: ball-query grouping on 64 keypoints ------------------ */
__global__ void group_kernel(const float* __restrict__ kp_src,
                             const float* __restrict__ kp_tgt,
                             float* __restrict__ dout)
{
    int tid = threadIdx.x;            /* 256 = 2 sets * 2 batches * 64 keypts */
    int set = tid >> 7;
    int b   = (tid >> 6) & 1;
    int s   = tid & 63;
    const float* kp = set ? kp_tgt : kp_src;
    float* outg = dout + (set ? OUT_TGTG : OUT_SRCG);
    const float* row = kp + ((size_t)b * KKEY + s) * 6;
    float cx = row[0], cy = row[1], cz = row[2];
    int cnt = 0, first = 0;
    bool have_first = false;
    for (int j = 0; j < KKEY; ++j) {
        const float* rj = kp + ((size_t)b * KKEY + j) * 6;
        float dx = rj[0] - cx, dy = rj[1] - cy, dz = rj[2] - cz;
        float d2 = dx * dx + dy * dy + dz * dz;
        if (d2 <= RADIUS2) {
            if (!have_first) { first = j; have_first = true; }
            if (cnt < NSAMP) {
                float* o = outg + (((size_t)(b * KKEY + s)) * NSAMP + cnt) * 6;
                o[0] = dx; o[1] = dy; o[2] = dz; o[3] = rj[3]; o[4] = rj[4]; o[5] = rj[5];
                ++cnt;
            }
        }
    }
    const float* rf = kp + ((size_t)b * KKEY + first) * 6;
    float fx = rf[0] - cx, fy = rf[1] - cy, fz = rf[2] - cz;
    for (; cnt < NSAMP; ++cnt) {
        float* o = outg + (((size_t)(b * KKEY + s)) * NSAMP + cnt) * 6;
        o[0] = fx; o[1] = fy; o[2] = fz; o[3] = rf[3]; o[4] = rf[4]; o[5] = rf[5];
    }
}

/* -------- kernel 4: 32-NN distances via V_WMMA_F32_16X16X4_F32 (exact f32) ----
 * d2 = A(16x4) * B(4x16) + C :  A row m = (qx,qy,qz,1)
 *                               B col n = (-2tx,-2ty,-2tz,|t|^2)
 *                               C[m][n] = |q_m|^2
 */
__global__ void __launch_bounds__(256)
knn_kernel(const float* __restrict__ cand,   /* (B, KKEY, NCAND, 3) */
           const float* __restrict__ tgt,    /* (B, 6, NPTS)        */
           float* __restrict__ dout)
{
    __shared__ float4 qs[QPB];                    /* x,y,z,|q|^2      */
    __shared__ float4 ts[TCHUNK];                 /* -2x,-2y,-2z,|t|^2 */
    __shared__ float  d2s[QPB * (TCHUNK + 1)];    /* padded stride 65 */
    __shared__ float  top[KNN * QPB];             /* column layout    */

    const int tid  = threadIdx.x;
    const int lane = tid & 31;
    const int wave = tid >> 5;
    const int mrow = lane & 15;
    const int hi   = lane >> 4;                   /* 0 or 1 */
    const int b    = blockIdx.y;
    const int qi   = blockIdx.x * QPB + tid;      /* always in range  */

    const float* qp = cand + ((size_t)b * NQ + qi) * 3;
    float qx = qp[0], qy = qp[1], qz = qp[2];
    qs[tid] = make_float4(qx, qy, qz, qx * qx + qy * qy + qz * qz);
#pragma unroll
    for (int i = 0; i < KNN; ++i) top[i * QPB + tid] = __builtin_inff();
    __syncthreads();

    /* loop-invariant A / C fragments for this wave's 2 row groups */
    v2f afrag[2];
    v8f cfrag[2];
#pragma unroll
    for (int g = 0; g < 2; ++g) {
        float4 qv = qs[wave * 32 + g * 16 + mrow];
        afrag[g].x = hi ? qv.z : qv.x;            /* K0|K2 in VGPR0 */
        afrag[g].y = hi ? 1.0f : qv.y;            /* K1|K3 in VGPR1 */
#pragma unroll
        for (int v = 0; v < 8; ++v)
            cfrag[g][v] = qs[wave * 32 + g * 16 + v + 8 * hi].w;
    }

    const float* tx = tgt + (size_t)b * 6 * NPTS;
    float worst = __builtin_inff();

    for (int ch = 0; ch < NCHUNK; ++ch) {
        __syncthreads();                          /* d2s/ts reuse fence */
        if (tid < TCHUNK) {
            int n = ch * TCHUNK + tid;
            float x = tx[n], y = tx[NPTS + n], z = tx[2 * NPTS + n];
            ts[tid] = make_float4(-2.f * x, -2.f * y, -2.f * z, x * x + y * y + z * z);
        }
        __syncthreads();

        v2f bfrag[4];
#pragma unroll
        for (int t = 0; t < 4; ++t) {
            float4 tv = ts[t * 16 + mrow];
            bfrag[t].x = hi ? tv.y : tv.x;        /* VGPR0: K0 (lo) / K1 (hi) */
            bfrag[t].y = hi ? tv.w : tv.z;        /* VGPR1: K2 (lo) / K3 (hi) */
        }
#pragma unroll
        for (int g = 0; g < 2; ++g) {
#pragma unroll
            for (int t = 0; t < 4; ++t) {
                v8f d = __builtin_amdgcn_wmma_f32_16x16x4_f32(
                    false, afrag[g], false, bfrag[t], (short)0, cfrag[g], false, false);
#pragma unroll
                for (int v = 0; v < 8; ++v) {
                    int r = wave * 32 + g * 16 + v + 8 * hi;
                    d2s[r * (TCHUNK + 1) + t * 16 + mrow] = d[v];
                }
            }
        }
        __syncthreads();

        /* per-thread top-32 insertion over this chunk's 64 candidates */
        for (int j = 0; j < TCHUNK; ++j) {
            float vv = d2s[tid * (TCHUNK + 1) + j];
            if (vv < worst) {
                int i = KNN - 1;
                while (i > 0 && top[(i - 1) * QPB + tid] > vv) {
                    top[i * QPB + tid] = top[(i - 1) * QPB + tid];
                    --i;
                }
                top[i * QPB + tid] = vv;
                worst = top[(KNN - 1) * QPB + tid];
            }
        }
    }

    float dist[KNN];
    float sum = 0.f;
#pragma unroll
    for (int i = 0; i < KNN; ++i) {
        dist[i] = sqrtf(fmaxf(top[i * QPB + tid], 0.f));
        sum += dist[i];
    }
    float inv = 1.0f / sum;
    float* o = dout + OUT_DIST + ((size_t)b * NQ + qi) * KNN;
#pragma unroll
    for (int i = 0; i < KNN; ++i) o[i] = dist[i] * inv;
}

/* ------------------------------- launcher ------------------------------------ */
extern "C" void kernel_launch(void* const* d_in, const int* in_sizes, int n_in,
                              void* d_out, int out_size, void* d_ws, size_t ws_size,
                              hipStream_t stream)
{
    const float* src_pts = (const float*)d_in[0];
    const float* tgt_pts = (const float*)d_in[1];
    const float* cand    = (const float*)d_in[2];
    const float* fe_w1   = (const float*)d_in[3];
    const float* fe_b1   = (const float*)d_in[4];
    const float* fe_w2   = (const float*)d_in[5];
    const float* fe_b2   = (const float*)d_in[6];
    const float* wl_w1   = (const float*)d_in[7];
    const float* wl_b1   = (const float*)d_in[8];
    const float* wl_w2   = (const float*)d_in[9];
    const float* wl_b2   = (const float*)d_in[10];
    float* out = (float*)d_out;
    float* ws  = (float*)d_ws;

    float* sc_src = ws;                 /* 2048 */
    float* sc_tgt = ws + 2048;          /* 2048 */
    float* kp_src = ws + 4096;          /* 768  */
    float* kp_tgt = ws + 4096 + 768;    /* 768  */

    score_kernel<<<NPTS / 256, 256, 0, stream>>>(src_pts, fe_w1, fe_b1, fe_w2, fe_b2,
                                                 wl_w1, wl_b1, wl_w2, wl_b2, sc_src);
    score_kernel<<<NPTS / 256, 256, 0, stream>>>(tgt_pts, fe_w1, fe_b1, fe_w2, fe_b2,
                                                 wl_w1, wl_b1, wl_w2, wl_b2, sc_tgt);
    topk_gather_kernel<<<1, 256, 0, stream>>>(src_pts, sc_src, kp_src, out + OUT_KEYPTS);
    topk_gather_kernel<<<1, 256, 0, stream>>>(tgt_pts, sc_tgt, kp_tgt, nullptr);
    group_kernel<<<1, 256, 0, stream>>>(kp_src, kp_tgt, out);
    knn_kernel<<<dim3(NQ / QPB, BATCH), 256, 0, stream>>>(cand, tgt_pts, out);
}